// Rank1ProjectorLoss_69346541961883
// MI455X (gfx1250) — compile-verified
//
#include <hip/hip_runtime.h>
#include <stdint.h>

typedef float v2f __attribute__((ext_vector_type(2)));
typedef float v8f __attribute__((ext_vector_type(8)));

#define ROWS   65536
#define COLS   1024
#define TILE_M 16
#define KC     32                 // columns per LDS chunk
#define PITCH  36                 // floats per LDS row (32 data + 4 pad -> bank-conflict-free A fetch)
#define NCHUNK (COLS / KC)        // 32
#define WPB    4                  // waves per block
#define THREADS (WPB * 32)
#define NBLOCKS 1024              // 4096 waves -> one 16-row tile per wave

// wait until ASYNCcnt <= N (async global->LDS loads complete in order)
#define WAIT_ASYNC(N) asm volatile("s_wait_asynccnt " #N ::: "memory")

// gfx1250 async DMA: global memory -> LDS, 16B per lane, tracked by ASYNCcnt.
// Data is streamed once (512 MiB > 192 MB L2): non-temporal hint avoids L2 churn.
__device__ __forceinline__ void async_b128(unsigned lds_off, const float* g) {
    asm volatile("global_load_async_to_lds_b128 %0, %1, off th:TH_LOAD_NT"
                 :: "v"(lds_off), "v"((unsigned long long)(uintptr_t)g)
                 : "memory");
}

__global__ void init_out_kernel(float* out) { out[0] = 0.0f; }

__global__ void __launch_bounds__(THREADS)
rank1_loss_kernel(const float* __restrict__ yp, const float* __restrict__ yt,
                  float* __restrict__ out) {
    // [wave][parity][matrix(P/T)][16 rows * PITCH]
    __shared__ float sh[WPB][2][2][TILE_M * PITCH];

    const int lane   = threadIdx.x & 31;
    const int wv     = threadIdx.x >> 5;
    const int gwave  = blockIdx.x * WPB + wv;
    const int nwaves = gridDim.x * WPB;
    const int ntiles = ROWS / TILE_M;

    // workgroup-relative LDS byte offsets of this wave's 4 buffers
    unsigned ldsB[2][2];
#pragma unroll
    for (int p = 0; p < 2; ++p)
#pragma unroll
        for (int m = 0; m < 2; ++m)
            ldsB[p][m] = (unsigned)(uintptr_t)(&sh[wv][p][m][0]);

    const int j0   = lane;               // float4 index base for async copy mapping
    const float EPSV = 1e-8f;
    const v2f   bones = {1.0f, 1.0f};    // B = ones(4x16): D[m][n] = sum_k A[m][k]
    const int   m_a = lane & 15;         // A-operand row for this lane (ISA 7.12.2)
    const int   kh  = lane >> 4;         // 0 -> K={0,1}, 1 -> K={2,3}

    for (int tile = gwave; tile < ntiles; tile += nwaves) {
        const float* gp = yp + (long)tile * TILE_M * COLS;
        const float* gt = yt + (long)tile * TILE_M * COLS;

        // prefetch chunk 0 into parity 0 (8 async b128 ops)
#pragma unroll
        for (int i = 0; i < 4; ++i) {
            int j = i * 32 + j0, r = j >> 3, cq = j & 7;
            unsigned lo = (unsigned)(r * PITCH * 4 + cq * 16);
            async_b128(ldsB[0][0] + lo, gp + r * COLS + cq * 4);
            async_b128(ldsB[0][1] + lo, gt + r * COLS + cq * 4);
        }

        v8f cpp = {}; v8f ctt = {}; v8f cpt = {};

        for (int c = 0; c < NCHUNK; ++c) {
            const int par = c & 1;
            if (c + 1 < NCHUNK) {
                const int kc = (c + 1) * KC, np = (c + 1) & 1;
#pragma unroll
                for (int i = 0; i < 4; ++i) {
                    int j = i * 32 + j0, r = j >> 3, cq = j & 7;
                    unsigned lo = (unsigned)(r * PITCH * 4 + cq * 16);
                    async_b128(ldsB[np][0] + lo, gp + r * COLS + kc + cq * 4);
                    async_b128(ldsB[np][1] + lo, gt + r * COLS + kc + cq * 4);
                }
                WAIT_ASYNC(8);   // chunk c's 8 transfers done (in-order), c+1 in flight
            } else {
                WAIT_ASYNC(0);
            }

            const float* bp = &sh[wv][par][0][0];
            const float* bt = &sh[wv][par][1][0];
#pragma unroll
            for (int k0 = 0; k0 < KC; k0 += 4) {
                const int off = m_a * PITCH + k0 + 2 * kh;   // 8B aligned, conflict-free banks
                v2f p = *(const v2f*)(bp + off);
                v2f t = *(const v2f*)(bt + off);
                p += EPSV; t += EPSV;
                v2f app = p * p, att = t * t, apt = p * t;
                cpp = __builtin_amdgcn_wmma_f32_16x16x4_f32(false, app, false, bones,
                                                            (short)0, cpp, false, false);
                ctt = __builtin_amdgcn_wmma_f32_16x16x4_f32(false, att, false, bones,
                                                            (short)0, ctt, false, false);
                cpt = __builtin_amdgcn_wmma_f32_16x16x4_f32(false, apt, false, bones,
                                                            (short)0, cpt, false, false);
            }
        }

        // D layout: lane<16 -> rows 0..7 in c[0..7]; lane>=16 -> rows 8..15.
        // Row sums replicated across N, so lanes 0 and 16 carry all 16 rows.
        if ((lane & 15) == 0) {
            float acc = 0.0f;
#pragma unroll
            for (int v = 0; v < 8; ++v) {
                float spp = cpp[v], stt = ctt[v], spt = cpt[v];
                acc += 2.0f - 2.0f * (spt * spt) / (spp * stt);
            }
            atomicAdd(out, acc * (1.0f / (float)ROWS));
        }
    }
}

extern "C" void kernel_launch(void* const* d_in, const int* in_sizes, int n_in,
                              void* d_out, int out_size, void* d_ws, size_t ws_size,
                              hipStream_t stream) {
    (void)in_sizes; (void)n_in; (void)out_size; (void)d_ws; (void)ws_size;
    const float* y_pred = (const float*)d_in[0];
    const float* y_true = (const float*)d_in[1];
    float* out = (float*)d_out;

    init_out_kernel<<<1, 1, 0, stream>>>(out);
    rank1_loss_kernel<<<NBLOCKS, THREADS, 0, stream>>>(y_pred, y_true, out);
}